// MGUModule_3667902071423
// MI455X (gfx1250) — compile-verified
//
#include <hip/hip_runtime.h>

#define S_LEN 2048
#define HDIM  2048
#define KDIM  2048
#define NWG   64
#define TPB   256
#define NWAVES (NWG * (TPB / 32))   // 512 waves in the persistent scan

typedef float v2f __attribute__((ext_vector_type(2)));
typedef float v8f __attribute__((ext_vector_type(8)));

// -------------------------------------------------------------------------
// Phase 1: out[S,H] = X[S,K] @ W[H,K]^T + bias  via V_WMMA_F32_16X16X4_F32
// Each wave owns one 16x16 output tile. Fragment layouts per ISA 7.12.2:
//   A (16x4 f32): lane l -> row m = l&15, holds K = kb + 2*(l>>4) and +1
//   B (4x16 f32): lane l -> col n = l&15 (== W row), same K packing
//   C/D: VGPR r, lanes 0-15 -> M=r, lanes 16-31 -> M=8+r, N = lane&15
// -------------------------------------------------------------------------
__global__ __launch_bounds__(256) void mgu_proj_gemm(
    const float* __restrict__ X, const float* __restrict__ W,
    const float* __restrict__ bias, float* __restrict__ out) {
  const int lane = threadIdx.x & 31;
  const int wave = threadIdx.x >> 5;
  const int gw   = blockIdx.x * 8 + wave;
  const int tiles_n = HDIM >> 4;                 // 128
  const int mt = gw / tiles_n;
  const int nt = gw - mt * tiles_n;
  const int m0 = mt << 4, n0 = nt << 4;
  const int r16 = lane & 15;
  const int kh  = (lane >> 4) << 1;              // 0 or 2
  const float* xr = X + (size_t)(m0 + r16) * KDIM + kh;
  const float* wr = W + (size_t)(n0 + r16) * KDIM + kh;

  v8f c0 = {};
  v8f c1 = {};
  for (int kb = 0; kb < KDIM; kb += 8) {         // two independent WMMA chains
    v2f a0 = *(const v2f*)(xr + kb);
    v2f b0 = *(const v2f*)(wr + kb);
    v2f a1 = *(const v2f*)(xr + kb + 4);
    v2f b1 = *(const v2f*)(wr + kb + 4);
    c0 = __builtin_amdgcn_wmma_f32_16x16x4_f32(false, a0, false, b0,
                                               (short)0, c0, false, false);
    c1 = __builtin_amdgcn_wmma_f32_16x16x4_f32(false, a1, false, b1,
                                               (short)0, c1, false, false);
  }
  v8f c = c0 + c1;

  const int n = n0 + r16;
  const float bv = bias[n];
  const int mbase = m0 + ((lane >> 4) << 3);
#pragma unroll
  for (int r = 0; r < 8; ++r)
    out[(size_t)(mbase + r) * HDIM + n] = c[r] + bv;
}

// -------------------------------------------------------------------------
// Grid-wide barrier: monotonic arrival counter, per-block epoch.
// Counter is zeroed by mgu_init each launch -> deterministic graph replay.
// -------------------------------------------------------------------------
__device__ __forceinline__ void grid_barrier(unsigned* bar, unsigned& epoch) {
  __threadfence();
  __syncthreads();
  ++epoch;
  if (threadIdx.x == 0) {
    __hip_atomic_fetch_add(bar, 1u, __ATOMIC_RELEASE, __HIP_MEMORY_SCOPE_AGENT);
    const unsigned goal = epoch * (unsigned)NWG;
    while (__hip_atomic_load(bar, __ATOMIC_ACQUIRE, __HIP_MEMORY_SCOPE_AGENT) < goal)
      __builtin_amdgcn_s_sleep(1);
  }
  __syncthreads();
}

__global__ void mgu_init(unsigned* bar) { *bar = 0u; }

// -------------------------------------------------------------------------
// Phase 2: persistent MGU scan. Weights (32 MB) live in the 192 MB L2.
// Wave-per-row GEMV, float4 coalesced, shfl_xor reduction (wave32).
// Two grid barriers per timestep (f depends on all of h; h_hat on all of f*h).
// -------------------------------------------------------------------------
__global__ __launch_bounds__(TPB) void mgu_scan(
    const float* __restrict__ Wfh, const float* __restrict__ bfh,
    const float* __restrict__ Whf, const float* __restrict__ bhf,
    const float* __restrict__ XF,  const float* __restrict__ XH,
    const float* __restrict__ h0,
    float* __restrict__ y,            // [S*H] then h_final[H]
    float* __restrict__ hbuf, float* __restrict__ fbuf,
    float* __restrict__ gbuf, unsigned* __restrict__ bar) {
  const int lane    = threadIdx.x & 31;
  const int wave_id = blockIdx.x * (TPB >> 5) + (threadIdx.x >> 5);
  unsigned epoch = 0;

  // h <- h0
  for (int i = blockIdx.x * TPB + threadIdx.x; i < HDIM; i += NWG * TPB)
    hbuf[i] = h0[i];
  grid_barrier(bar, epoch);

  for (int t = 0; t < S_LEN; ++t) {
    const float* xf_t = XF + (size_t)t * HDIM;
    // --- Phase A: f = sigmoid(xf_t + Wfh@h + bfh); g = f*h ---
    for (int r = wave_id; r < HDIM; r += NWAVES) {
      const float* wr = Wfh + (size_t)r * HDIM;
      if (r + NWAVES < HDIM)
        __builtin_prefetch(Wfh + (size_t)(r + NWAVES) * HDIM + (lane << 2), 0, 0);
      float4 acc = make_float4(0.f, 0.f, 0.f, 0.f);
      for (int k = lane << 2; k < HDIM; k += 128) {
        const float4 w  = *(const float4*)(wr + k);
        const float4 hv = *(const float4*)(hbuf + k);
        acc.x = fmaf(w.x, hv.x, acc.x);
        acc.y = fmaf(w.y, hv.y, acc.y);
        acc.z = fmaf(w.z, hv.z, acc.z);
        acc.w = fmaf(w.w, hv.w, acc.w);
      }
      float s = (acc.x + acc.y) + (acc.z + acc.w);
#pragma unroll
      for (int off = 16; off > 0; off >>= 1) s += __shfl_xor(s, off, 32);
      if (lane == 0) {
        const float z = xf_t[r] + s + bfh[r];
        const float f = 1.0f / (1.0f + __expf(-z));
        fbuf[r] = f;
        gbuf[r] = f * hbuf[r];
      }
    }
    grid_barrier(bar, epoch);

    // --- Phase B: h_hat = tanh(Whf@g + bhf + xh_t); h = (1-f)h + f*h_hat ---
    const float* xh_t = XH + (size_t)t * HDIM;
    float* yt = y + (size_t)t * HDIM;
    for (int r = wave_id; r < HDIM; r += NWAVES) {
      const float* wr = Whf + (size_t)r * HDIM;
      if (r + NWAVES < HDIM)
        __builtin_prefetch(Whf + (size_t)(r + NWAVES) * HDIM + (lane << 2), 0, 0);
      float4 acc = make_float4(0.f, 0.f, 0.f, 0.f);
      for (int k = lane << 2; k < HDIM; k += 128) {
        const float4 w  = *(const float4*)(wr + k);
        const float4 gv = *(const float4*)(gbuf + k);
        acc.x = fmaf(w.x, gv.x, acc.x);
        acc.y = fmaf(w.y, gv.y, acc.y);
        acc.z = fmaf(w.z, gv.z, acc.z);
        acc.w = fmaf(w.w, gv.w, acc.w);
      }
      float s = (acc.x + acc.y) + (acc.z + acc.w);
#pragma unroll
      for (int off = 16; off > 0; off >>= 1) s += __shfl_xor(s, off, 32);
      if (lane == 0) {
        const float hh = tanhf(s + bhf[r] + xh_t[r]);
        const float f  = fbuf[r];
        const float ho = hbuf[r];
        const float hn = fmaf(f, hh - ho, ho);   // (1-f)*h + f*h_hat
        hbuf[r] = hn;
        yt[r]   = hn;
      }
    }
    grid_barrier(bar, epoch);
  }

  // h_final appended after y
  for (int i = blockIdx.x * TPB + threadIdx.x; i < HDIM; i += NWG * TPB)
    y[(size_t)S_LEN * HDIM + i] = hbuf[i];
}

// -------------------------------------------------------------------------
extern "C" void kernel_launch(void* const* d_in, const int* in_sizes, int n_in,
                              void* d_out, int out_size, void* d_ws, size_t ws_size,
                              hipStream_t stream) {
  const float* x   = (const float*)d_in[0];
  const float* h   = (const float*)d_in[1];
  const float* Wfx = (const float*)d_in[2];
  const float* bfx = (const float*)d_in[3];
  const float* Wfh = (const float*)d_in[4];
  const float* bfh = (const float*)d_in[5];
  const float* Whf = (const float*)d_in[6];
  const float* bhf = (const float*)d_in[7];
  const float* Whx = (const float*)d_in[8];
  const float* bhx = (const float*)d_in[9];
  float* y = (float*)d_out;

  char* ws = (char*)d_ws;
  unsigned* bar = (unsigned*)ws;                         // 256B control slot
  float* XF   = (float*)(ws + 256);                      // [S*H]
  float* XH   = XF + (size_t)S_LEN * HDIM;               // [S*H]
  float* hbuf = XH + (size_t)S_LEN * HDIM;               // [H]
  float* fbuf = hbuf + HDIM;                             // [H]
  float* gbuf = fbuf + HDIM;                             // [H]

  hipLaunchKernelGGL(mgu_init, dim3(1), dim3(1), 0, stream, bar);

  dim3 gGemm((S_LEN / 16) * (HDIM / 16) / 8);            // 2048 blocks, wave/tile
  hipLaunchKernelGGL(mgu_proj_gemm, gGemm, dim3(256), 0, stream, x, Wfx, bfx, XF);
  hipLaunchKernelGGL(mgu_proj_gemm, gGemm, dim3(256), 0, stream, x, Whx, bhx, XH);

  hipLaunchKernelGGL(mgu_scan, dim3(NWG), dim3(TPB), 0, stream,
                     Wfh, bfh, Whf, bhf, XF, XH, h, y, hbuf, fbuf, gbuf, bar);
}